// SpectralFeatureExtractor_63582695850469
// MI455X (gfx1250) — compile-verified
//
#include <hip/hip_runtime.h>
#include <hip/hip_bf16.h>
#include <math.h>

// ---------------------------------------------------------------------------
// SpectralFeatureExtractor for MI455X (gfx1250, wave32)
//   Stage 1: 24-pt rDFT (bins 1..12) -> top-2 magnitude -> amp/phase feats
//   Stage 2: fused fp32 MLP via V_WMMA_F32_16X16X4_F32
// ---------------------------------------------------------------------------

typedef __attribute__((ext_vector_type(2))) float v2f;
typedef __attribute__((ext_vector_type(8))) float v8f;

#define BATCH 4096
#define CHANS 64
#define TLEN  256
#define NFFT  24
#define FBINS 12
#define DIN   256    // C * TOPK * 2
#define FEAT  512
#define TILE_M 16

// cos/sin(2*pi*m/24), m = 0..23  (15-degree steps). Fully constant-folded
// into immediates by the unrolled DFT below.
__device__ constexpr float COS24[24] = {
     1.0000000000f,  0.9659258263f,  0.8660254038f,  0.7071067812f,
     0.5000000000f,  0.2588190451f,  0.0000000000f, -0.2588190451f,
    -0.5000000000f, -0.7071067812f, -0.8660254038f, -0.9659258263f,
    -1.0000000000f, -0.9659258263f, -0.8660254038f, -0.7071067812f,
    -0.5000000000f, -0.2588190451f,  0.0000000000f,  0.2588190451f,
     0.5000000000f,  0.7071067812f,  0.8660254038f,  0.9659258263f
};
__device__ constexpr float SIN24[24] = {
     0.0000000000f,  0.2588190451f,  0.5000000000f,  0.7071067812f,
     0.8660254038f,  0.9659258263f,  1.0000000000f,  0.9659258263f,
     0.8660254038f,  0.7071067812f,  0.5000000000f,  0.2588190451f,
     0.0000000000f, -0.2588190451f, -0.5000000000f, -0.7071067812f,
    -0.8660254038f, -0.9659258263f, -1.0000000000f, -0.9659258263f,
    -0.8660254038f, -0.7071067812f, -0.5000000000f, -0.2588190451f
};

// ---------------------------------------------------------------------------
// Stage 1: one thread per (b, c) row. Reads x[b][c][0:24], writes
// feats[b][c*4 + {amp0, amp1, ph0, ph1}] laid out flat as [B, 256].
// ---------------------------------------------------------------------------
__global__ __launch_bounds__(256)
void feat_kernel(const float* __restrict__ x, float* __restrict__ feats)
{
    const int r = blockIdx.x * blockDim.x + threadIdx.x;   // 0 .. B*C-1
    const float* px = x + (size_t)r * TLEN;

    float s[NFFT];
#pragma unroll
    for (int i = 0; i < NFFT / 4; ++i) {
        float4 v = *(const float4*)(px + i * 4);
        s[i * 4 + 0] = v.x; s[i * 4 + 1] = v.y;
        s[i * 4 + 2] = v.z; s[i * 4 + 3] = v.w;
    }

    float re[FBINS], im[FBINS], mag[FBINS];
#pragma unroll
    for (int j = 0; j < FBINS; ++j) {
        const int k = j + 1;                    // FFT bin (skip DC)
        float sr = 0.0f, si = 0.0f;
#pragma unroll
        for (int t = 0; t < NFFT; ++t) {
            const int m = (k * t) % 24;         // compile-time constant
            sr = fmaf(s[t],  COS24[m], sr);
            si = fmaf(s[t], -SIN24[m], si);     // exp(-i*theta)
        }
        re[j] = sr;
        im[j] = si;
        mag[j] = sqrtf(fmaf(sr, sr, si * si));
    }

    // top-2 (descending, ties -> lower index), branch-free cndmask chains
    float best = -1.0f, bre = 0.0f, bim = 0.0f; int bi = 0;
#pragma unroll
    for (int j = 0; j < FBINS; ++j) {
        bool t = mag[j] > best;
        best = t ? mag[j] : best;
        bre  = t ? re[j]  : bre;
        bim  = t ? im[j]  : bim;
        bi   = t ? j      : bi;
    }
    float sec = -1.0f, sre = 0.0f, sim = 0.0f;
#pragma unroll
    for (int j = 0; j < FBINS; ++j) {
        bool t = (j != bi) && (mag[j] > sec);
        sec = t ? mag[j] : sec;
        sre = t ? re[j]  : sre;
        sim = t ? im[j]  : sim;
    }

    float4 o;
    o.x = best;
    o.y = sec;
    o.z = atan2f(bim, bre);
    o.w = atan2f(sim, sre);
    *(float4*)(feats + (size_t)r * 4) = o;
}

// ---------------------------------------------------------------------------
// Stage 2: fused MLP. 256 threads = 8 waves; block computes 16 batch rows.
// Wave w owns output columns [w*64, w*64+64) = 4 WMMA N-tiles.
//
// WMMA f32 16x16x4 operand striping (wave32):
//   A: lane l -> row m = l&15, VGPR0 = A[m][kb], VGPR1 = A[m][kb+1],
//      kb = 2*(l>>4)
//   B: lane l -> col n = l&15, VGPR0 = B[kb][n], VGPR1 = B[kb+1][n]
//   C/D: lane l, vgpr v -> (M = v + 8*(l>>4), N = l&15)
// ---------------------------------------------------------------------------
__global__ __launch_bounds__(256)
void mlp_kernel(const float* __restrict__ feats,
                const float* __restrict__ W1, const float* __restrict__ b1,
                const float* __restrict__ W2, const float* __restrict__ b2,
                float* __restrict__ out)
{
    constexpr int FSTRIDE = DIN + 4;    // 260: break 64-bank alignment
    constexpr int HSTRIDE = FEAT + 4;   // 516
    __shared__ float sF[TILE_M * FSTRIDE];   // 16.25 KB
    __shared__ float sH[TILE_M * HSTRIDE];   // 32.25 KB

    const int tid     = threadIdx.x;
    const int wave    = tid >> 5;
    const int lane    = tid & 31;
    const int rowBase = blockIdx.x * TILE_M;

    // ---- cooperative load of feats tile [16 x 256] into padded LDS
    {
        const float4* src = (const float4*)(feats + (size_t)rowBase * DIN);
#pragma unroll
        for (int i = 0; i < 4; ++i) {
            const int v4  = tid + i * 256;        // float4 index 0..1023
            float4 v      = src[v4];
            const int lin = v4 * 4;               // element index
            const int r   = lin >> 8;             // /256
            const int c   = lin & 255;
            *(float4*)(&sF[r * FSTRIDE + c]) = v;
        }
    }
    __syncthreads();

    const int m    = lane & 15;            // A row within tile
    const int n    = lane & 15;            // B/D column within N-tile
    const int kb   = (lane >> 4) << 1;     // 0 or 2
    const int half = lane >> 4;            // 0/1 -> D rows v or v+8
    const int nBase = wave * 64;

    // ---------------- GEMM1: h = gelu(feats @ W1 + b1) ----------------
    v8f acc[4] = {};
    for (int k = 0; k < DIN / 4; ++k) {
        v2f a = *(const v2f*)(&sF[m * FSTRIDE + k * 4 + kb]);
#pragma unroll
        for (int t = 0; t < 4; ++t) {
            const int col = nBase + t * 16 + n;
            v2f b;
            b.x = W1[(size_t)(k * 4 + kb)     * FEAT + col];
            b.y = W1[(size_t)(k * 4 + kb + 1) * FEAT + col];
            acc[t] = __builtin_amdgcn_wmma_f32_16x16x4_f32(
                false, a, false, b, (short)0, acc[t], false, false);
        }
    }

    // bias + exact-erf GELU -> LDS h tile
#pragma unroll
    for (int t = 0; t < 4; ++t) {
        const int col = nBase + t * 16 + n;
        const float bias = b1[col];
#pragma unroll
        for (int v = 0; v < 8; ++v) {
            const int row = v + 8 * half;
            float hx = acc[t][v] + bias;
            float g  = 0.5f * hx * (1.0f + erff(hx * 0.7071067811865476f));
            sH[row * HSTRIDE + col] = g;
        }
    }
    __syncthreads();

    // ---------------- GEMM2: out = h @ W2 + b2 ----------------
    v8f acc2[4] = {};
    for (int k = 0; k < FEAT / 4; ++k) {
        v2f a = *(const v2f*)(&sH[m * HSTRIDE + k * 4 + kb]);
#pragma unroll
        for (int t = 0; t < 4; ++t) {
            const int col = nBase + t * 16 + n;
            v2f b;
            b.x = W2[(size_t)(k * 4 + kb)     * FEAT + col];
            b.y = W2[(size_t)(k * 4 + kb + 1) * FEAT + col];
            acc2[t] = __builtin_amdgcn_wmma_f32_16x16x4_f32(
                false, a, false, b, (short)0, acc2[t], false, false);
        }
    }

#pragma unroll
    for (int t = 0; t < 4; ++t) {
        const int col = nBase + t * 16 + n;
        const float bias = b2[col];
#pragma unroll
        for (int v = 0; v < 8; ++v) {
            const int row = v + 8 * half;
            out[(size_t)(rowBase + row) * FEAT + col] = acc2[t][v] + bias;
        }
    }
}

// ---------------------------------------------------------------------------
// Host-side launch. Inputs (setup_inputs order): x, W1, b1, W2, b2.
// d_ws holds the intermediate feats [4096 x 256] f32 (4 MB).
// ---------------------------------------------------------------------------
extern "C" void kernel_launch(void* const* d_in, const int* in_sizes, int n_in,
                              void* d_out, int out_size, void* d_ws, size_t ws_size,
                              hipStream_t stream)
{
    (void)in_sizes; (void)n_in; (void)out_size; (void)ws_size;

    const float* x  = (const float*)d_in[0];
    const float* W1 = (const float*)d_in[1];
    const float* b1 = (const float*)d_in[2];
    const float* W2 = (const float*)d_in[3];
    const float* b2 = (const float*)d_in[4];
    float* outp  = (float*)d_out;
    float* feats = (float*)d_ws;

    // Stage 1: 262144 rows, 256 threads/block
    feat_kernel<<<(BATCH * CHANS) / 256, 256, 0, stream>>>(x, feats);

    // Stage 2: 256 blocks x 256 threads, 16 batch rows per block
    mlp_kernel<<<BATCH / TILE_M, 256, 0, stream>>>(feats, W1, b1, W2, b2, outp);
}